// GraphNets_89593017794874
// MI455X (gfx1250) — compile-verified
//
#include <hip/hip_runtime.h>

#define BSZ     512
#define NNODES  22
#define INDIM   128
#define HID     256
#define ROWS    (BSZ * NNODES)   // 11264, divisible by 16

typedef __attribute__((ext_vector_type(2))) float v2f;
typedef __attribute__((ext_vector_type(8))) float v8f;

// Exact-fp32 tensor op: D(16x16) = A(16x4) x B(4x16) + C
__device__ __forceinline__ v8f wmma4(v2f a, v2f b, v8f c) {
    return __builtin_amdgcn_wmma_f32_16x16x4_f32(
        /*neg_a=*/false, a, /*neg_b=*/false, b,
        /*c_mod=*/(short)0, c, /*reuse_a=*/false, /*reuse_b=*/false);
}

// K0: W1s[k][n] = We1[k][n] + We1[k+128][n]   (both halves of edge_feats are x[a])
__global__ void gn_prep_w1s(const float* __restrict__ We1, float* __restrict__ W1s) {
    int i = blockIdx.x * blockDim.x + threadIdx.x;
    if (i < INDIM * HID) W1s[i] = We1[i] + We1[INDIM * HID + i];
}

// K1: per 16-row tile of the 11264 distinct edge rows:
//     H1 = ReLU(X @ W1s + be1) (K=128) -> LDS;  E = H1 @ We2 + be2 (K=256)
__global__ void __launch_bounds__(128)
gn_edge(const float* __restrict__ X,    // [ROWS,128]
        const float* __restrict__ W1s,  // [128,256]
        const float* __restrict__ be1,  // [256]
        const float* __restrict__ We2,  // [256,256]
        const float* __restrict__ be2,  // [256]
        float* __restrict__ E)          // [ROWS,256]
{
    __shared__ float h1[16][HID + 4];
    const int row0 = blockIdx.x * 16;
    const int tid  = threadIdx.x;
    const int wave = tid >> 5;
    const int lane = tid & 31;
    const int m    = lane & 15;          // A row / B,C,D column-in-tile
    const int kg   = (lane >> 4) << 1;   // K sub-offset: 0 or 2
    const int rsel = (lane >> 4) << 3;   // C/D row offset: 0 or 8

    // ---- stage 1: H1 tile (16 x 256), wave owns 64 columns ----
    const float* xrow = X + (size_t)(row0 + m) * INDIM;
    for (int nt = 0; nt < 4; ++nt) {
        const int col = wave * 64 + nt * 16 + m;
        v8f acc = {};
        for (int kb = 0; kb < INDIM; kb += 4) {
            v2f a = { xrow[kb + kg], xrow[kb + kg + 1] };
            v2f b = { W1s[(kb + kg) * HID + col], W1s[(kb + kg + 1) * HID + col] };
            acc = wmma4(a, b, acc);
        }
        const float bias = be1[col];
#pragma unroll
        for (int v = 0; v < 8; ++v)
            h1[v + rsel][col] = fmaxf(acc[v] + bias, 0.0f);
    }
    __syncthreads();

    // ---- stage 2: E tile = H1 @ We2 + be2 (K = 256) ----
    for (int nt = 0; nt < 4; ++nt) {
        const int col = wave * 64 + nt * 16 + m;
        v8f acc = {};
        for (int kb = 0; kb < HID; kb += 4) {
            v2f a = { h1[m][kb + kg], h1[m][kb + kg + 1] };
            v2f b = { We2[(kb + kg) * HID + col], We2[(kb + kg + 1) * HID + col] };
            acc = wmma4(a, b, acc);
        }
        const float bias = be2[col];
#pragma unroll
        for (int v = 0; v < 8; ++v)
            E[(size_t)(row0 + v + rsel) * HID + col] = acc[v] + bias;
    }
}

// K2: agg[s][c] = sum_{a<22} E[s*22+a][c]   (fixed order -> deterministic)
__global__ void gn_reduce(const float* __restrict__ E, float* __restrict__ agg) {
    int i = blockIdx.x * blockDim.x + threadIdx.x;
    if (i < BSZ * HID) {
        const int s = i / HID, c = i % HID;
        const float* p = E + (size_t)s * NNODES * HID + c;
        float sum = 0.0f;
#pragma unroll
        for (int a = 0; a < NNODES; ++a) sum += p[a * HID];
        agg[i] = sum;
    }
}

// K3: node MLP on 512 rows, broadcast each result row to all 22 node slots.
__global__ void __launch_bounds__(128)
gn_node(const float* __restrict__ AGG,  // [512,256]
        const float* __restrict__ Wn1, const float* __restrict__ bn1,
        const float* __restrict__ Wn2, const float* __restrict__ bn2,
        float* __restrict__ OUT)        // [512,22,256]
{
    __shared__ float h2[16][HID + 4];
    const int row0 = blockIdx.x * 16;
    const int tid  = threadIdx.x;
    const int wave = tid >> 5;
    const int lane = tid & 31;
    const int m    = lane & 15;
    const int kg   = (lane >> 4) << 1;
    const int rsel = (lane >> 4) << 3;

    const float* arow = AGG + (size_t)(row0 + m) * HID;
    for (int nt = 0; nt < 4; ++nt) {
        const int col = wave * 64 + nt * 16 + m;
        v8f acc = {};
        for (int kb = 0; kb < HID; kb += 4) {
            v2f a = { arow[kb + kg], arow[kb + kg + 1] };
            v2f b = { Wn1[(kb + kg) * HID + col], Wn1[(kb + kg + 1) * HID + col] };
            acc = wmma4(a, b, acc);
        }
        const float bias = bn1[col];
#pragma unroll
        for (int v = 0; v < 8; ++v)
            h2[v + rsel][col] = fmaxf(acc[v] + bias, 0.0f);
    }
    __syncthreads();

    for (int nt = 0; nt < 4; ++nt) {
        const int col = wave * 64 + nt * 16 + m;
        v8f acc = {};
        for (int kb = 0; kb < HID; kb += 4) {
            v2f a = { h2[m][kb + kg], h2[m][kb + kg + 1] };
            v2f b = { Wn2[(kb + kg) * HID + col], Wn2[(kb + kg + 1) * HID + col] };
            acc = wmma4(a, b, acc);
        }
        const float bias = bn2[col];
#pragma unroll
        for (int v = 0; v < 8; ++v) {
            const int   s   = row0 + v + rsel;          // frame index
            const float val = acc[v] + bias;
            float* o = OUT + (size_t)s * NNODES * HID + col;
#pragma unroll
            for (int b = 0; b < NNODES; ++b) o[(size_t)b * HID] = val;
        }
    }
}

extern "C" void kernel_launch(void* const* d_in, const int* in_sizes, int n_in,
                              void* d_out, int out_size, void* d_ws, size_t ws_size,
                              hipStream_t stream) {
    const float* x   = (const float*)d_in[0];
    const float* We1 = (const float*)d_in[1];
    const float* be1 = (const float*)d_in[2];
    const float* We2 = (const float*)d_in[3];
    const float* be2 = (const float*)d_in[4];
    const float* Wn1 = (const float*)d_in[5];
    const float* bn1 = (const float*)d_in[6];
    const float* Wn2 = (const float*)d_in[7];
    const float* bn2 = (const float*)d_in[8];
    float* out = (float*)d_out;

    // workspace layout (floats): W1s[32768] | agg[131072] | E[11264*256]
    float* W1s = (float*)d_ws;
    float* agg = W1s + INDIM * HID;
    float* E   = agg + BSZ * HID;

    gn_prep_w1s<<<(INDIM * HID + 255) / 256, 256, 0, stream>>>(We1, W1s);
    gn_edge<<<ROWS / 16, 128, 0, stream>>>(x, W1s, be1, We2, be2, E);
    gn_reduce<<<(BSZ * HID + 255) / 256, 256, 0, stream>>>(E, agg);
    gn_node<<<BSZ / 16, 128, 0, stream>>>(agg, Wn1, bn1, Wn2, bn2, out);
}